// ALTER2Layer_16020228014696
// MI455X (gfx1250) — compile-verified
//
#include <hip/hip_runtime.h>

typedef float v2f __attribute__((ext_vector_type(2)));
typedef float v8f __attribute__((ext_vector_type(8)));

// Explicit global-address-space pointers so every inner-loop access lowers to
// global_load/global_store (never flat_*, which would tie up the LDS path and
// DScnt). C-style casts across address spaces are supported by HIP clang.
using gcf = const __attribute__((address_space(1))) float;
using gf  = __attribute__((address_space(1))) float;
using gv2 = const __attribute__((address_space(1))) v2f;

// Generic f32 WMMA GEMM over batch dim z:  C[z] = post( (k-scaled A[z]) @ B[z] )
// TA: A element (m,k) at A[k*lda + m]   (A stored [K,M] row-major)
// TB: B element (k,n) at B[n*ldb + k]   (B stored [N,K] row-major)
// KS: multiply A element (m,k) by ks[k] (per-batch vector)
// EPI: 0 = store raw
//      1 = val * rs[m] * cs[n]
//      2 = sigmoid(val + bias[n])
//      3 = EPI2 + deriv[m,n] = val*(1-val)
// RM: M-tiles per wave. Block = 128 threads = 4 waves; block tile
// (RM*64)(M) x 64(N); each wave owns RM x 4 16x16 tiles (4*RM WMMAs / K-step).
// M multiple of RM*64, N multiple of 64, K multiple of 4.
template<bool TA, bool TB, bool KS, int EPI, int RM>
__global__ __launch_bounds__(128)
void wmma_gemm_f32(const float* __restrict__ Ap, long sAb, int lda,
                   const float* __restrict__ Bp, long sBb, int ldb,
                   float* __restrict__ Cp, long sCb, long ldc,
                   const float* __restrict__ ksp, long sKb,
                   const float* __restrict__ rsp, long sRb,
                   const float* __restrict__ csp, long sCsb,
                   const float* __restrict__ bias,
                   float* __restrict__ dp, long sDb, long ldd,
                   int K)
{
  const int z = blockIdx.z;
  gcf* A = (gcf*)(Ap + (long)z * sAb);
  gcf* B = (gcf*)(Bp + (long)z * sBb);
  gf*  C = (gf*)(Cp + (long)z * sCb);

  const int lane = threadIdx.x & 31;
  const int wave = threadIdx.x >> 5;
  const int lm   = lane & 15;    // M (A) / N (B,C) index within a 16-tile
  const int lh   = lane >> 4;    // K-half for A/B frags, M-half for C/D
  const int m0   = blockIdx.y * (RM * 64) + wave * (RM * 16);
  const int n0   = blockIdx.x * 64;

  v8f acc[RM][4] = {};

  // Strength-reduced fragment pointers (constant stride per K-step).
  gcf* pA0 = nullptr; gcf* pA1 = nullptr;   // TA: rows k, k+1 (coalesced)
  gcf* pAr[RM] = {};                        // !TA: one row pointer per M-tile
  if constexpr (TA) {
    pA0 = A + (long)(2 * lh) * lda + (m0 + lm);
    pA1 = pA0 + lda;
  } else {
    #pragma unroll
    for (int r = 0; r < RM; ++r)
      pAr[r] = A + (long)(m0 + r * 16 + lm) * lda + 2 * lh;
  }
  gcf* pBt[4] = {};                         // TB: one col pointer per N-tile
  gcf* pB0 = nullptr; gcf* pB1 = nullptr;   // !TB: rows k, k+1 (coalesced)
  if constexpr (TB) {
    #pragma unroll
    for (int t = 0; t < 4; ++t)
      pBt[t] = B + (long)(n0 + t * 16 + lm) * ldb + 2 * lh;
  } else {
    pB0 = B + (long)(2 * lh) * ldb + (n0 + lm);
    pB1 = pB0 + ldb;
  }
  gcf* pk = nullptr;
  if constexpr (KS) pk = (gcf*)(ksp + (long)z * sKb) + 2 * lh;

  for (int k = 0; k < K; k += 4) {
    v2f av[RM];
    if constexpr (TA) {
      #pragma unroll
      for (int r = 0; r < RM; ++r) { av[r].x = pA0[r * 16]; av[r].y = pA1[r * 16]; }
      pA0 += 4 * (long)lda; pA1 += 4 * (long)lda;
    } else {
      #pragma unroll
      for (int r = 0; r < RM; ++r) { av[r] = *(gv2*)pAr[r]; pAr[r] += 4; }
    }
    if constexpr (KS) {
      const float k0 = pk[0], k1 = pk[1];  pk += 4;
      #pragma unroll
      for (int r = 0; r < RM; ++r) { av[r].x *= k0; av[r].y *= k1; }
    }
    v2f bv[4];
    if constexpr (TB) {
      #pragma unroll
      for (int t = 0; t < 4; ++t) { bv[t] = *(gv2*)pBt[t]; pBt[t] += 4; }
    } else {
      #pragma unroll
      for (int t = 0; t < 4; ++t) { bv[t].x = pB0[t * 16]; bv[t].y = pB1[t * 16]; }
      pB0 += 4 * (long)ldb; pB1 += 4 * (long)ldb;
    }
    #pragma unroll
    for (int r = 0; r < RM; ++r)
      #pragma unroll
      for (int t = 0; t < 4; ++t)
        acc[r][t] = __builtin_amdgcn_wmma_f32_16x16x4_f32(
            false, av[r], false, bv[t], (short)0, acc[r][t], false, false);
  }

  // Epilogue. C/D layout: VGPR v -> M = mbase + v + 8*lh, N = n0 + t*16 + lm.
  gcf* rs = nullptr; gcf* cs = nullptr; gf* Dv = nullptr; gcf* bi = nullptr;
  if constexpr (EPI == 1) {
    rs = (gcf*)(rsp + (long)z * sRb);
    cs = (gcf*)(csp + (long)z * sCsb);
  }
  if constexpr (EPI >= 2) bi = (gcf*)bias;
  if constexpr (EPI == 3) Dv = (gf*)(dp + (long)z * sDb);

  #pragma unroll
  for (int r = 0; r < RM; ++r) {
    const int mb = m0 + r * 16 + 8 * lh;
    float rsv[8];
    if constexpr (EPI == 1) {
      #pragma unroll
      for (int v = 0; v < 8; ++v) rsv[v] = rs[mb + v];
    }
    #pragma unroll
    for (int t = 0; t < 4; ++t) {
      const int n = n0 + t * 16 + lm;
      float cst = 1.f, badd = 0.f;
      if constexpr (EPI == 1) cst = cs[n];
      if constexpr (EPI >= 2) badd = bi[n];
      #pragma unroll
      for (int v = 0; v < 8; ++v) {
        const long m = mb + v;
        float val = acc[r][t][v];
        if constexpr (EPI == 1) val = val * rsv[v] * cst;
        if constexpr (EPI >= 2) val = 1.f / (1.f + __expf(-(val + badd)));
        C[m * ldc + n] = val;
        if constexpr (EPI == 3) Dv[m * ldd + n] = val * (1.f - val);
      }
    }
  }
}

extern "C" void kernel_launch(void* const* d_in, const int* in_sizes, int n_in,
                              void* d_out, int out_size, void* d_ws, size_t ws_size,
                              hipStream_t stream)
{
  (void)in_sizes; (void)n_in; (void)out_size;
  const float* x  = (const float*)d_in[0];   // [128,1024]
  const float* W1 = (const float*)d_in[1];   // [512,1024]
  const float* b1 = (const float*)d_in[2];   // [512]
  const float* W2 = (const float*)d_in[3];   // [256,512]
  const float* b2 = (const float*)d_in[4];   // [256]
  const float* b3 = (const float*)d_in[5];   // [512]
  const float* br = (const float*)d_in[6];   // [1024]
  float* out = (float*)d_out;
  float* ws  = (float*)d_ws;

  const int B = 128, D = 1024, H1 = 512, H2 = 256;

  // d_out layout: recover [128,1024], c2 [128,256], Jac flat as (d, b, e)
  float* recover = out;
  float* c2      = out + (long)B * D;                 // 131072
  float* jac     = out + (long)B * D + (long)B * H2;  // 163840

  // Workspace layout (floats)
  float* c1 = ws;                    // 65536
  float* s1 = ws + 65536;            // 65536
  float* s2 = ws + 131072;           // 32768
  float* c3 = ws + 163840;           // 65536
  float* s3 = ws + 229376;           // 65536
  const long wsFixed = 294912;
  const long midSz = (long)H1 * H1;  // 262144 per batch
  const long qSz   = (long)H1 * D;   // 524288 per batch
  int chunk = 16;
  while (chunk > 1 &&
         (size_t)(wsFixed + (long)chunk * (midSz + qSz)) * sizeof(float) > ws_size)
    chunk >>= 1;
  float* mid = ws + wsFixed;
  float* q   = mid + (long)chunk * midSz;

  dim3 blk(128, 1, 1);

  // --- Forward pass (RM=2: block tile 128x64 matches B=128) ---
  // c1 = sigmoid(x @ W1^T + b1); s1 = c1*(1-c1)
  wmma_gemm_f32<false, true, false, 3, 2><<<dim3(H1/64, 1, 1), blk, 0, stream>>>(
      x, 0L, D,   W1, 0L, D,   c1, 0L, (long)H1,
      nullptr, 0L, nullptr, 0L, nullptr, 0L, b1, s1, 0L, (long)H1, D);
  // c2 = sigmoid(c1 @ W2^T + b2); s2
  wmma_gemm_f32<false, true, false, 3, 2><<<dim3(H2/64, 1, 1), blk, 0, stream>>>(
      c1, 0L, H1,  W2, 0L, H1,  c2, 0L, (long)H2,
      nullptr, 0L, nullptr, 0L, nullptr, 0L, b2, s2, 0L, (long)H2, H1);
  // c3 = sigmoid(c2 @ W2 + b3); s3
  wmma_gemm_f32<false, false, false, 3, 2><<<dim3(H1/64, 1, 1), blk, 0, stream>>>(
      c2, 0L, H2,  W2, 0L, H1,  c3, 0L, (long)H1,
      nullptr, 0L, nullptr, 0L, nullptr, 0L, b3, s3, 0L, (long)H1, H2);
  // recover = sigmoid(c3 @ W1 + b_r)
  wmma_gemm_f32<false, false, false, 2, 2><<<dim3(D/64, 1, 1), blk, 0, stream>>>(
      c3, 0L, H1,  W1, 0L, D,   recover, 0L, (long)D,
      nullptr, 0L, nullptr, 0L, nullptr, 0L, br, nullptr, 0L, 0L, H1);

  // --- Jacobian (RM=4: block tile 256x64, 16 WMMAs / K-step / wave) ---
  // Jst[b] = W1^T @ (diag(s1) W2^T diag(s2) W2 diag(s3)) @ W1
  for (int c0 = 0; c0 < B; c0 += chunk) {
    // MidS[b](i,j) = s1[i] * s3[j] * sum_e W2[e][i]*s2[e]*W2[e][j]
    wmma_gemm_f32<true, false, true, 1, 4><<<dim3(H1/64, H1/256, chunk), blk, 0, stream>>>(
        W2, 0L, H1,  W2, 0L, H1,  mid, midSz, (long)H1,
        s2 + (long)c0 * H2, (long)H2,
        s1 + (long)c0 * H1, (long)H1,
        s3 + (long)c0 * H1, (long)H1,
        nullptr, nullptr, 0L, 0L, H2);
    // Q[b] = MidS[b] @ W1
    wmma_gemm_f32<false, false, false, 0, 4><<<dim3(D/64, H1/256, chunk), blk, 0, stream>>>(
        mid, midSz, H1,  W1, 0L, D,  q, qSz, (long)D,
        nullptr, 0L, nullptr, 0L, nullptr, 0L,
        nullptr, nullptr, 0L, 0L, H1);
    // Jst[b] = W1^T @ Q[b], written at out_jac[d*B*D + b*D + e]
    wmma_gemm_f32<true, false, false, 0, 4><<<dim3(D/64, D/256, chunk), blk, 0, stream>>>(
        W1, 0L, D,   q, qSz, D,   jac + (long)c0 * D, (long)D, (long)B * D,
        nullptr, 0L, nullptr, 0L, nullptr, 0L,
        nullptr, nullptr, 0L, 0L, H1);
  }
}